// Embedding_for_one_hot_inputs_45835890983007
// MI455X (gfx1250) — compile-verified
//
#include <hip/hip_runtime.h>

// ---------------------------------------------------------------------------
// Embedding-via-one-hot, MI455X (gfx1250, wave32).
//
// out[b,s,:] = weight[argmax_v onehot[b,s,v], :]   (bit-exact vs dense einsum)
//
// Index extraction uses the matrix unit exactly:
//   id = sum_v onehot[v] * v   (every term/partial exact in f32, V < 2^24)
// done as v_wmma_f32_16x16x4_f32 with iota folded into A and B == 1.0
// (B layout independent). Memory-bound: ~137 MB total -> ~6 us @ 23.3 TB/s.
// ---------------------------------------------------------------------------

typedef __attribute__((ext_vector_type(2))) float v2f;
typedef __attribute__((ext_vector_type(4))) float v4f;
typedef __attribute__((ext_vector_type(8))) float v8f;

#define BB    8
#define SS    128
#define VV    32000
#define EE    768
#define ROWS  (BB * SS)        // 1024 (b,s) rows
#define KTILE 256              // K-span handled per block
#define NKT   (VV / KTILE)     // 125 K-tiles

// --- phase 0: zero the f32 index accumulators in workspace ------------------
__global__ void zero_idx_kernel(float* __restrict__ idxf) {
    int i = blockIdx.x * blockDim.x + threadIdx.x;
    if (i < ROWS) idxf[i] = 0.0f;
}

// --- phase 1: id[row] += sum_{k in tile} onehot[row][k] * k  via WMMA -------
// One wave (32 threads) per block; EXEC all-ones as WMMA requires.
// grid = (NKT, ROWS/16).  A-matrix 16x4 f32 layout (ISA 7.12.2):
//   lane L<16 : row M=L,    K = k0, k0+1   (VGPR0, VGPR1)
//   lane L>=16: row M=L-16, K = k0+2, k0+3
// C/D 16x16 f32 layout: VGPR g -> lanes 0-15 hold row M=g, lanes 16-31 M=g+8.
__global__ __launch_bounds__(32)
void onehot_argmax_wmma(const float* __restrict__ onehot,
                        float* __restrict__ idxf) {
    const int lane    = threadIdx.x;          // 0..31
    const int rowtile = blockIdx.y;           // 0..63
    const int ktile   = blockIdx.x;           // 0..124
    const int m       = lane & 15;
    const int kk      = (lane >> 4) << 1;     // 0 or 2

    const long rowbase = (long)(rowtile * 16 + m) * VV;
    int k = ktile * KTILE + kk;               // this lane's first global K

    const float* p = onehot + rowbase + k;

    v8f c = {};                               // f32 accumulator (8 VGPRs)
    const v2f bones = {1.0f, 1.0f};           // B == all-ones (layout agnostic)

    #pragma unroll 4
    for (int it = 0; it < KTILE / 4; ++it) {
        // stream one-hot non-temporally: 131 MB read exactly once
        v2f a = __builtin_nontemporal_load((const v2f*)p);
        a.x *= (float)k;                      // fold iota into A (exact)
        a.y *= (float)(k + 1);
        // D = A x B + C : per-row K-reduction on the matrix unit
        c = __builtin_amdgcn_wmma_f32_16x16x4_f32(
                /*neg_a=*/false, a, /*neg_b=*/false, bones,
                /*c_mod=*/(short)0, c, /*reuse_a=*/false, /*reuse_b=*/false);
        p += 4;
        k += 4;
    }

    // Column 0 of D holds the per-row partial index sums. Only one partial
    // across all K-tiles is nonzero, so f32 atomic adds are exact.
    if (lane == 0 || lane == 16) {
        const int rbase = rowtile * 16 + ((lane >> 4) << 3);
        #pragma unroll
        for (int g = 0; g < 8; ++g)
            atomicAdd(&idxf[rbase + g], c[g]);
    }
}

// --- phase 2: coalesced row gather: out[row,:] = weight[id[row],:] ----------
__global__ __launch_bounds__(192)
void gather_rows(const float* __restrict__ weight,
                 const float* __restrict__ idxf,
                 float* __restrict__ out) {
    const int row = blockIdx.x;               // 0..1023
    const int t   = threadIdx.x;              // 0..191 (192 * float4 = 768)
    const int idx = (int)(idxf[row] + 0.5f);  // exact integer in f32

    const v4f* __restrict__ wr = (const v4f*)(weight + (long)idx * EE);
    v4f*       __restrict__ o  = (v4f*)(out + (long)row * EE);
    o[t] = wr[t];
}

// ---------------------------------------------------------------------------
extern "C" void kernel_launch(void* const* d_in, const int* in_sizes, int n_in,
                              void* d_out, int out_size, void* d_ws, size_t ws_size,
                              hipStream_t stream) {
    const float* onehot = (const float*)d_in[0];  // [B,S,V] f32 one-hot
    const float* weight = (const float*)d_in[1];  // [V,E]   f32
    float*       out    = (float*)d_out;          // [B,S,E] f32
    float*       idxf   = (float*)d_ws;           // ROWS f32 accumulators

    zero_idx_kernel<<<(ROWS + 255) / 256, 256, 0, stream>>>(idxf);

    dim3 grid(NKT, ROWS / 16);                    // 125 x 64 single-wave blocks
    onehot_argmax_wmma<<<grid, 32, 0, stream>>>(onehot, idxf);

    gather_rows<<<ROWS, 192, 0, stream>>>(weight, idxf, out);
}